// WeightsLib2DBeta_71751723647104
// MI455X (gfx1250) — compile-verified
//
#include <hip/hip_runtime.h>

typedef __attribute__((ext_vector_type(2))) float v2f;
typedef __attribute__((ext_vector_type(8))) float v8f;

#define HW_POS      (1024u * 1024u)   // H*W positions
#define E_STRIDE_F2 (2u * HW_POS)     // float2 stride between w0 and w1 planes
#define C_STRIDE_F2 (HW_POS)          // float2 stride between channel 0 / 1
#define N_PAIRS     (HW_POS / 32u)    // 32768 pairs of 16-position tiles
#define EPSF        1e-12f

// ---------------------------------------------------------------------------
// Kernel 1: tiny MLP (x@W_in.T + b) * tanh(x@W_gate.T + b), then compose the
// clerp∘caddmul affine map into per-(batch,channel,component) 4-coefficient
// rows + bias. 16 threads, one per batch. Output: coef[16][2][2][4], bias[16][2][2].
// ---------------------------------------------------------------------------
__global__ void wl2d_params_kernel(const float* __restrict__ x,
                                   const float* __restrict__ W_in,
                                   const float* __restrict__ b_in,
                                   const float* __restrict__ W_gate,
                                   const float* __restrict__ b_gate,
                                   float* __restrict__ coef,
                                   float* __restrict__ bias)
{
    unsigned b = threadIdx.x;
    if (b >= 16u) return;

    float mod[12];
    for (int i = 0; i < 12; ++i) {
        float t = b_in[i], g = b_gate[i];
        for (int k = 0; k < 64; ++k) {
            float xv = x[b * 64u + k];
            t = fmaf(xv, W_in[i * 64 + k], t);
            g = fmaf(xv, W_gate[i * 64 + k], g);
        }
        mod[i] = t * tanhf(g);
    }

    for (int c = 0; c < 2; ++c) {
        float l0 = mod[0 * 4 + c * 2 + 0] + 0.5f;   // param_l + shift_lerp
        float l1 = mod[0 * 4 + c * 2 + 1];
        float b0 = mod[1 * 4 + c * 2 + 0];          // param_b
        float b1 = mod[1 * 4 + c * 2 + 1];
        float s0 = mod[2 * 4 + c * 2 + 0] + 1.0f;   // param_s + shift_scale
        float s1 = mod[2 * 4 + c * 2 + 1];

        // clerp coefficients over (w0re, w0im, w1re, w1im):
        //   z_re: (1-l0),  l1,      l0,  -l1
        //   z_im:   -l1,  (1-l0),   l1,   l0
        // caddmul: z'_re = b0 + s0*z_re - s1*z_im ; z'_im = b1 + s1*z_re + s0*z_im
        float cre[4] = { s0 * (1.f - l0) + s1 * l1,
                         s0 * l1 - s1 * (1.f - l0),
                         s0 * l0 - s1 * l1,
                        -s0 * l1 - s1 * l0 };
        float cim[4] = { s1 * (1.f - l0) - s0 * l1,
                         s1 * l1 + s0 * (1.f - l0),
                         s1 * l0 + s0 * l1,
                         s0 * l0 - s1 * l1 };

        float* cf0 = coef + (((b * 2u + c) * 2u + 0u) * 4u);
        float* cf1 = coef + (((b * 2u + c) * 2u + 1u) * 4u);
        for (int k = 0; k < 4; ++k) { cf0[k] = cre[k]; cf1[k] = cim[k]; }
        bias[(b * 2u + c) * 2u + 0u] = b0;
        bias[(b * 2u + c) * 2u + 1u] = b1;
    }
}

// ---------------------------------------------------------------------------
// Kernel 2: main sweep. Each wave processes PAIRS of 16-position tiles.
//   A (16x4, f32): rows m = 2*b_local + comp, per (batch-group g, channel c).
//       lanes 0-15 carry K0,K1 ; lanes 16-31 carry K2,K3  (per ISA A layout).
//   B (4x16, f32): K rows = (w0re,w0im,w1re,w1im), N = position in tile.
//       lanes 0-15 -> K0/K1 (w0 float2), lanes 16-31 -> K2/K3 (w1 float2):
//       one NT b64 load per lane per (channel, tile), no cross-lane shuffles.
//   C = per-row bias. D(16x16) = z' values; epilogue does the channel-1
//   rsqrt normalization + channel-0 dot, then NT b32 stores.
// ---------------------------------------------------------------------------
__global__ void wl2d_wmma_kernel(const float* __restrict__ weights,
                                 const float* __restrict__ coef,
                                 const float* __restrict__ bias,
                                 float* __restrict__ out)
{
    const unsigned lane  = threadIdx.x & 31u;
    const unsigned m     = lane & 15u;   // A-row / tile-column index
    const unsigned khalf = lane >> 4;    // 0: K0,K1 (w0)   1: K2,K3 (w1)

    // Build A and C fragments once per wave (uniform over the tile loop).
    v2f a[2][2];
    v8f cf[2][2];
    for (int g = 0; g < 2; ++g) {
        for (int c = 0; c < 2; ++c) {
            unsigned bb   = g * 8u + (m >> 1);
            unsigned comp = m & 1u;
            const float* cp = coef + (((bb * 2u + c) * 2u + comp) * 4u);
            v2f av;
            av.x = cp[2u * khalf + 0u];
            av.y = cp[2u * khalf + 1u];
            a[g][c] = av;
            for (int v = 0; v < 8; ++v) {
                unsigned row = (unsigned)v + (khalf ? 8u : 0u);   // D/C row map
                unsigned br  = g * 8u + (row >> 1);
                cf[g][c][v]  = bias[(br * 2u + c) * 2u + (row & 1u)];
            }
        }
    }

    const v2f* wf2 = (const v2f*)weights;
    const unsigned nWaves = (gridDim.x * blockDim.x) >> 5;
    const unsigned waveId = (blockIdx.x * blockDim.x + threadIdx.x) >> 5;

    for (unsigned tp = waveId; tp < N_PAIRS; tp += nWaves) {
        const unsigned pos0  = tp * 32u + m;        // tile 2*tp
        const unsigned pos1  = pos0 + 16u;          // tile 2*tp+1
        const unsigned base0 = khalf * E_STRIDE_F2 + pos0;
        const unsigned base1 = base0 + 16u;

        // Speculative prefetch of the next pair (dropped silently if OOB),
        // unconditional -> no EXEC bookkeeping in the loop.
        __builtin_prefetch(&wf2[base0 + nWaves * 32u], 0, 0);
        __builtin_prefetch(&wf2[base0 + C_STRIDE_F2 + nWaves * 32u], 0, 0);

        v2f w0c0 = __builtin_nontemporal_load(&wf2[base0]);
        v2f w0c1 = __builtin_nontemporal_load(&wf2[base0 + C_STRIDE_F2]);
        v2f w1c0 = __builtin_nontemporal_load(&wf2[base1]);
        v2f w1c1 = __builtin_nontemporal_load(&wf2[base1 + C_STRIDE_F2]);

        // 8x V_WMMA_F32_16X16X4_F32 : D = A*B + bias, back-to-back for XDL ILP
        v8f d00a = __builtin_amdgcn_wmma_f32_16x16x4_f32(false, a[0][0], false, w0c0,
                                                         (short)0, cf[0][0], false, false);
        v8f d01a = __builtin_amdgcn_wmma_f32_16x16x4_f32(false, a[0][1], false, w0c1,
                                                         (short)0, cf[0][1], false, false);
        v8f d10a = __builtin_amdgcn_wmma_f32_16x16x4_f32(false, a[1][0], false, w0c0,
                                                         (short)0, cf[1][0], false, false);
        v8f d11a = __builtin_amdgcn_wmma_f32_16x16x4_f32(false, a[1][1], false, w0c1,
                                                         (short)0, cf[1][1], false, false);
        v8f d00b = __builtin_amdgcn_wmma_f32_16x16x4_f32(false, a[0][0], false, w1c0,
                                                         (short)0, cf[0][0], false, false);
        v8f d01b = __builtin_amdgcn_wmma_f32_16x16x4_f32(false, a[0][1], false, w1c1,
                                                         (short)0, cf[0][1], false, false);
        v8f d10b = __builtin_amdgcn_wmma_f32_16x16x4_f32(false, a[1][0], false, w1c0,
                                                         (short)0, cf[1][0], false, false);
        v8f d11b = __builtin_amdgcn_wmma_f32_16x16x4_f32(false, a[1][1], false, w1c1,
                                                         (short)0, cf[1][1], false, false);

        // Epilogue: out[b,pos] = (z0 . z1) / sqrt(|z1|^2 + eps)
        // D row v (+8 for lanes>=16) = 2*b_local + comp; (re,im) adjacent.
        #pragma unroll
        for (int t = 0; t < 2; ++t) {
            const unsigned pos = t ? pos1 : pos0;
            #pragma unroll
            for (int g = 0; g < 2; ++g) {
                v8f z0 = t ? (g ? d10b : d00b) : (g ? d10a : d00a);
                v8f z1 = t ? (g ? d11b : d01b) : (g ? d11a : d01a);
                #pragma unroll
                for (int k = 0; k < 4; ++k) {
                    float d0r = z0[2 * k + 0], d0i = z0[2 * k + 1];
                    float d1r = z1[2 * k + 0], d1i = z1[2 * k + 1];
                    unsigned b = (unsigned)g * 8u + (unsigned)k + (khalf ? 4u : 0u);
                    float dot = fmaf(d0r, d1r, d0i * d1i);
                    float nrm = fmaf(d1r, d1r, fmaf(d1i, d1i, EPSF));
                    float r   = dot * rsqrtf(nrm);
                    __builtin_nontemporal_store(r, &out[b * HW_POS + pos]);
                }
            }
        }
    }
}

extern "C" void kernel_launch(void* const* d_in, const int* in_sizes, int n_in,
                              void* d_out, int out_size, void* d_ws, size_t ws_size,
                              hipStream_t stream)
{
    (void)in_sizes; (void)n_in; (void)out_size; (void)ws_size;

    const float* x       = (const float*)d_in[0];
    const float* W_in    = (const float*)d_in[1];
    const float* b_in    = (const float*)d_in[2];
    const float* W_gate  = (const float*)d_in[3];
    const float* b_gate  = (const float*)d_in[4];
    const float* weights = (const float*)d_in[5];
    float* out = (float*)d_out;

    float* coef = (float*)d_ws;     // 16*2*2*4 = 256 floats
    float* bias = coef + 256;       // 16*2*2   =  64 floats

    wl2d_params_kernel<<<1, 16, 0, stream>>>(x, W_in, b_in, W_gate, b_gate, coef, bias);

    // 32768 tile-pairs; 1024 blocks * 8 waves = 8192 waves -> 4 pairs/wave.
    wl2d_wmma_kernel<<<1024, 256, 0, stream>>>(weights, coef, bias, out);
}